// DeepSupervisionBoundaryDoULoss_6021544149054
// MI455X (gfx1250) — compile-verified
//
#include <hip/hip_runtime.h>
#include <hip/hip_bf16.h>

typedef __attribute__((ext_vector_type(2))) float v2f;
typedef __attribute__((ext_vector_type(8))) float v8f;

#define THREADS 256

__device__ __forceinline__ float sigmoidf_(float x) {
    return 1.0f / (1.0f + __expf(-x));
}

// One fused streaming pass for one deep-supervision scale.
// grid = 32 planes * bpp blocks-per-plane, 256 threads.
// Each block reduces its 5 partial sums (S, C, inter, z, psum) with
// V_WMMA_F32_16X16X4_F32 and writes them to a private workspace slot
// (deterministic: no float atomics).
__global__ __launch_bounds__(THREADS)
void bdou_scale_kernel(const float* __restrict__ logits,
                       const int* __restrict__ targets,
                       float* __restrict__ partial,
                       int shift, int h, int w, int w4log, int bpp)
{
    const int plane = blockIdx.x / bpp;          // 0..31  (b*N + n)
    const int bp    = blockIdx.x - plane * bpp;  // block-in-plane
    const int tid   = threadIdx.x;
    const int W4    = w >> 2;

    const float* L = logits + (long)plane * h * w;
    const int*   T = targets + (long)plane * 512 * 512;  // full-res targets

    float accS = 0.f, accC = 0.f, accI = 0.f, accZ = 0.f, accP = 0.f;

    const int P4 = h * W4;  // float4 chunks per plane
    for (int idx = bp * THREADS + tid; idx < P4; idx += bpp * THREADS) {
        const int y  = idx >> w4log;
        const int x0 = (idx - (y << w4log)) << 2;

        const float4 lg = *(const float4*)(L + (long)y * w + x0);
        float p[4] = { sigmoidf_(lg.x), sigmoidf_(lg.y),
                       sigmoidf_(lg.z), sigmoidf_(lg.w) };

        const int* Trow = T + (y << shift) * 512;   // nearest-DS row
        int t[4];
#pragma unroll
        for (int j = 0; j < 4; ++j) t[j] = (Trow[(x0 + j) << shift] != 0);

#pragma unroll
        for (int j = 0; j < 4; ++j) {
            accP += p[j];
            accZ += p[j] * p[j];
            if (t[j]) {
                accS += 1.0f;
                accI += p[j];
                const int x = x0 + j;
                // 3x3 cross: interior iff all 4 neighbors are 1 (zero pad)
                int tu = (y > 0)     ? (T[((y - 1) << shift) * 512 + (x << shift)] != 0) : 0;
                int td = (y < h - 1) ? (T[((y + 1) << shift) * 512 + (x << shift)] != 0) : 0;
                int tl = (x > 0)     ? (Trow[(x - 1) << shift] != 0) : 0;
                int tr = (x < w - 1) ? (Trow[(x + 1) << shift] != 0) : 0;
                if (!(tu & td & tl & tr)) accC += 1.0f;
            }
        }
    }

    __shared__ float red[5][THREADS];
    red[0][tid] = accS; red[1][tid] = accC; red[2][tid] = accI;
    red[3][tid] = accZ; red[4][tid] = accP;
    __syncthreads();

    if (tid < 32) {   // wave 0 only: EXEC all-ones inside (no divergence)
        const int lane = tid;
#if defined(__has_builtin)
#if __has_builtin(__builtin_amdgcn_wmma_f32_16x16x4_f32)
#define BDOU_HAVE_WMMA 1
#endif
#endif
#ifdef BDOU_HAVE_WMMA
        // Reduce 256 floats: four chained 16x4 A-matrices x all-ones B,
        // C-accumulated.  Exact f32 FMA arithmetic.
        const v2f ones = {1.0f, 1.0f};
#pragma unroll
        for (int q = 0; q < 5; ++q) {
            v8f c = {0.f, 0.f, 0.f, 0.f, 0.f, 0.f, 0.f, 0.f};
#pragma unroll
            for (int ch = 0; ch < 4; ++ch) {
                v2f a;
                a.x = red[q][ch * 64 + 2 * lane];
                a.y = red[q][ch * 64 + 2 * lane + 1];
                // 8 args: (neg_a, A, neg_b, B, c_mod, C, reuse_a, reuse_b)
                c = __builtin_amdgcn_wmma_f32_16x16x4_f32(
                        false, a, false, ones, (short)0, c, false, false);
            }
            // C/D layout: lane L (<16) holds D[v][L], lane L+16 holds D[v+8][L].
            // sum_v c[v]  +  partner half  ==  full column sum == total.
            float s = c[0] + c[1] + c[2] + c[3] + c[4] + c[5] + c[6] + c[7];
            s += __shfl_xor(s, 16);
            if (lane == 0) partial[blockIdx.x * 5 + q] = s;
        }
#else
        for (int q = 0; q < 5; ++q) {
            float s = 0.f;
            for (int i = lane; i < THREADS; i += 32) s += red[q][i];
            for (int off = 16; off > 0; off >>= 1) s += __shfl_xor(s, off);
            if (lane == 0) partial[blockIdx.x * 5 + q] = s;
        }
#endif
    }
}

// Single-wave finalize: per-plane alpha/dou, valid-masked mean, scale weights.
__global__ void bdou_finalize(const float* __restrict__ p0,
                              const float* __restrict__ p1,
                              const float* __restrict__ p2,
                              const float* __restrict__ valid_mask,
                              float* __restrict__ out)
{
    const int pl = threadIdx.x;  // 0..31 plane id
    const float* parts[3] = { p0, p1, p2 };
    const int   bpps[3]  = { 32, 8, 2 };
    const float invhw[3] = { 1.0f / (512.f * 512.f),
                             1.0f / (256.f * 256.f),
                             1.0f / (128.f * 128.f) };
    const float wgt[3] = { 4.0f / 7.0f, 2.0f / 7.0f, 1.0f / 7.0f };

    float valid = (valid_mask[pl] >= 0.5f) ? 1.0f : 0.0f;
    float cnt = valid;
    for (int off = 16; off > 0; off >>= 1) cnt += __shfl_xor(cnt, off);

    float loss = 0.0f;
    for (int s = 0; s < 3; ++s) {
        float S = 0.f, C = 0.f, I = 0.f, Z = 0.f, P = 0.f;
        const float* base = parts[s] + pl * bpps[s] * 5;
        for (int b = 0; b < bpps[s]; ++b) {
            S += base[b * 5 + 0]; C += base[b * 5 + 1]; I += base[b * 5 + 2];
            Z += base[b * 5 + 3]; P += base[b * 5 + 4];
        }
        float per;
        if (S > 0.0f) {
            float alpha = 2.0f * (1.0f - (C + 1e-5f) / (S + 1e-5f)) - 1.0f;
            alpha = fminf(alpha, 0.8f);
            // y_sum == S for binary targets
            per = (Z + S - 2.0f * I + 1e-5f) /
                  (Z + S - (1.0f + alpha) * I + 1e-5f);
        } else {
            per = P * invhw[s];  // empty-target fallback: mean(probs)
        }
        float v = per * valid;
        for (int off = 16; off > 0; off >>= 1) v += __shfl_xor(v, off);
        loss += wgt[s] * ((cnt > 0.0f) ? (v / cnt) : 0.0f);
    }
    if (pl == 0) out[0] = loss;
}

extern "C" void kernel_launch(void* const* d_in, const int* in_sizes, int n_in,
                              void* d_out, int out_size, void* d_ws, size_t ws_size,
                              hipStream_t stream) {
    (void)in_sizes; (void)n_in; (void)out_size; (void)ws_size;
    const float* logits0 = (const float*)d_in[0];   // [8,4,512,512] f32
    const float* logits1 = (const float*)d_in[1];   // [8,4,256,256] f32
    const float* logits2 = (const float*)d_in[2];   // [8,4,128,128] f32
    const int*   targets = (const int*)d_in[3];     // [8,4,512,512] i32
    const float* valid   = (const float*)d_in[4];   // [8,4] f32
    float* out = (float*)d_out;

    float* ws = (float*)d_ws;
    float* p0 = ws;                 // 32 planes * 32 blocks * 5 = 5120 floats
    float* p1 = ws + 32 * 32 * 5;   // 32 * 8 * 5 = 1280 floats
    float* p2 = p1 + 32 * 8 * 5;    // 32 * 2 * 5 = 320 floats

    bdou_scale_kernel<<<32 * 32, THREADS, 0, stream>>>(logits0, targets, p0, 0, 512, 512, 7, 32);
    bdou_scale_kernel<<<32 * 8,  THREADS, 0, stream>>>(logits1, targets, p1, 1, 256, 256, 6, 8);
    bdou_scale_kernel<<<32 * 2,  THREADS, 0, stream>>>(logits2, targets, p2, 2, 128, 128, 5, 2);
    bdou_finalize<<<1, 32, 0, stream>>>(p0, p1, p2, valid, out);
}